// Log2_Int_Quantizer_42004780154947
// MI455X (gfx1250) — compile-verified
//
#include <hip/hip_runtime.h>
#include <stdint.h>

// Log2 integer quantizer for attention maps, MI455X (gfx1250).
//
// Reference semantics collapse to:
//   e(x)  = floor(log2(xi)) + (xi >= 1.5*2^floor(log2 xi)),  xi = round(x/s)
//   dq    = 2^e   (xi==0 -> dq=0 sentinel)
//   out   = clip( floor( dq*256/2^emax ) [floor(./2) if #unique>16], 0, 255 ) * (s*255)
// All global state reduces to a presence mask of e-codes (bits 0..31) + zero bit 33.
//
// Memory-bound problem (zero matrix FLOPs): fast path traffic = 250 MB
// (read x 100 + write codes 25 | read codes 25 + write out 100) ~= 10.7 us at 23.3 TB/s.

#define ZERO_CODE 33u

__device__ __forceinline__ unsigned log_code(float xh, float s) {
  int xi = (int)rintf(xh / s);            // exact IEEE div + RNE, matches jnp.round
  if (xi <= 0) return ZERO_CODE;          // zero sentinel path
  int l = 31 - __clz(xi);                 // floor(log2)
  int e = l;
  if (l > 0 && (unsigned)xi >= (3u << (l - 1))) e += 1;  // frac >= 0.5 in log2 domain
  return (unsigned)e;                     // e in [0,31]
}

__device__ __forceinline__ float decode1(unsigned c, int shift, float s_out) {
  if (c == ZERO_CODE) return 0.0f;        // dq=0 -> out 0
  int k = (int)c + shift;                 // shift = 8 - emax - halve; k <= 8 since e <= emax
  if (k < 0) return 0.0f;                 // floor of sub-1 power of two
  float p = __uint_as_float((unsigned)(127 + k) << 23);  // exact 2^k
  return fminf(p, 255.0f) * s_out;        // clip (256 -> 255)
}

__device__ __forceinline__ float dequant1(float xh, float s, int shift, float s_out) {
  return decode1(log_code(xh, s), shift, s_out);
}

__global__ void init_mask_kernel(unsigned long long* mask) { *mask = 0ull; }

// ---------------- Pass 1 ----------------
// Variant A (fast path): stream x, write packed u8 codes, reduce presence mask.
__global__ __launch_bounds__(256) void quant_mask_store_kernel(
    const float* __restrict__ x, const float* __restrict__ s_ptr,
    unsigned long long* __restrict__ mask_out, unsigned char* __restrict__ codes,
    long long n4, long long n) {
  __shared__ unsigned long long lmask;
  if (threadIdx.x == 0) lmask = 0ull;
  __syncthreads();
  const float s = *s_ptr;
  unsigned long long m = 0ull;
  const float4* __restrict__ x4 = (const float4*)x;
  unsigned* __restrict__ cw = (unsigned*)codes;
  const long long stride = (long long)gridDim.x * 256;
  const long long tid = (long long)blockIdx.x * 256 + threadIdx.x;
  for (long long i = tid; i < n4; i += stride) {
    if (i + stride < n4) __builtin_prefetch(&x4[i + stride], 0, 0);  // global_prefetch_b8
    float4 v = x4[i];
    unsigned c0 = log_code(v.x, s), c1 = log_code(v.y, s);
    unsigned c2 = log_code(v.z, s), c3 = log_code(v.w, s);
    m |= (1ull << c0) | (1ull << c1) | (1ull << c2) | (1ull << c3);
    cw[i] = c0 | (c1 << 8) | (c2 << 16) | (c3 << 24);
  }
  for (long long i = n4 * 4 + tid; i < n; i += stride) {
    unsigned c = log_code(x[i], s);
    m |= 1ull << c;
    codes[i] = (unsigned char)c;
  }
  atomicOr(&lmask, m);                    // ds_or_b64
  __syncthreads();
  if (threadIdx.x == 0) atomicOr(mask_out, lmask);  // global_atomic_or_b64
}

// Variant B (fallback, no code buffer): mask only.
__global__ __launch_bounds__(256) void quant_mask_kernel(
    const float* __restrict__ x, const float* __restrict__ s_ptr,
    unsigned long long* __restrict__ mask_out, long long n4, long long n) {
  __shared__ unsigned long long lmask;
  if (threadIdx.x == 0) lmask = 0ull;
  __syncthreads();
  const float s = *s_ptr;
  unsigned long long m = 0ull;
  const float4* __restrict__ x4 = (const float4*)x;
  const long long stride = (long long)gridDim.x * 256;
  const long long tid = (long long)blockIdx.x * 256 + threadIdx.x;
  for (long long i = tid; i < n4; i += stride) {
    if (i + stride < n4) __builtin_prefetch(&x4[i + stride], 0, 0);
    float4 v = x4[i];
    m |= (1ull << log_code(v.x, s)) | (1ull << log_code(v.y, s)) |
         (1ull << log_code(v.z, s)) | (1ull << log_code(v.w, s));
  }
  for (long long i = n4 * 4 + tid; i < n; i += stride)
    m |= 1ull << log_code(x[i], s);
  atomicOr(&lmask, m);
  __syncthreads();
  if (threadIdx.x == 0) atomicOr(mask_out, lmask);
}

// Derive (emax, halve) from the mask; emit s_out scalar output.
__global__ void finalize_kernel(const unsigned long long* __restrict__ mask,
                                const float* __restrict__ s_ptr,
                                int* __restrict__ params,
                                float* __restrict__ s_out_slot) {
  unsigned long long m = *mask;
  unsigned ebits = (unsigned)(m & 0xFFFFFFFFull);
  int anyzero = (int)((m >> ZERO_CODE) & 1ull);
  int emax = ebits ? (31 - __clz(ebits)) : 0;
  int lo = emax - 8;                                    // e < lo collapses to out==0
  unsigned lowm = (lo > 0) ? ((1u << lo) - 1u) : 0u;
  int nuniq = __popc(ebits & ~lowm) + ((anyzero || (ebits & lowm)) ? 1 : 0);
  int halve = (nuniq > 16) ? 1 : 0;
  params[0] = emax;
  params[1] = halve;
  *s_out_slot = *s_ptr * 255.0f;
}

// ---------------- Pass 2, fast path ----------------
// Decode packed u8 codes -> floats. Codes staged through LDS with a
// double-buffered async-to-LDS DMA pipeline (16 codes / lane / b128).
__global__ __launch_bounds__(256) void dequant_codes_kernel(
    const unsigned char* __restrict__ codes, const float* __restrict__ s_ptr,
    const int* __restrict__ params, float* __restrict__ out,
    long long n4, long long n) {
#if defined(__gfx1250__)
  __shared__ uint4 tile[2][256];
#endif
  const float s_out = (*s_ptr) * 255.0f;
  const int shift = 8 - params[0] - params[1];
  const unsigned* __restrict__ cw = (const unsigned*)codes;
  float4* __restrict__ out4 = (float4*)out;
  const long long stride = gridDim.x;
#if defined(__gfx1250__)
  const long long nchunks = n4 >> 10;     // chunk = 1024 code-words (4 KB LDS, 16 KB out)
  const uint4* __restrict__ cq = (const uint4*)codes;
  long long c = blockIdx.x;
  if (c < nchunks) {
    {  // prologue: chunk c -> buffer 0
      unsigned laddr = (unsigned)(uintptr_t)(void*)&tile[0][threadIdx.x];
      unsigned long long gaddr = (unsigned long long)(cq + c * 256 + threadIdx.x);
      asm volatile("global_load_async_to_lds_b128 %0, %1, off"
                   :: "v"(laddr), "v"(gaddr) : "memory");
    }
    int buf = 0;
    for (;;) {
      long long cn = c + stride;
      if (cn < nchunks) {  // keep one chunk in flight
        unsigned laddr = (unsigned)(uintptr_t)(void*)&tile[buf ^ 1][threadIdx.x];
        unsigned long long gaddr = (unsigned long long)(cq + cn * 256 + threadIdx.x);
        asm volatile("global_load_async_to_lds_b128 %0, %1, off"
                     :: "v"(laddr), "v"(gaddr) : "memory");
        asm volatile("s_wait_asynccnt 0x1" ::: "memory");
      } else {
        asm volatile("s_wait_asynccnt 0x0" ::: "memory");
      }
      uint4 w = tile[buf][threadIdx.x];    // ds_load_b128: 16 codes
      const long long wbase = c * 1024 + (long long)threadIdx.x * 4;
      unsigned ws_[4] = {w.x, w.y, w.z, w.w};
#pragma unroll
      for (int j = 0; j < 4; ++j) {
        unsigned wj = ws_[j];
        float4 r;
        r.x = decode1(wj & 0xFFu, shift, s_out);
        r.y = decode1((wj >> 8) & 0xFFu, shift, s_out);
        r.z = decode1((wj >> 16) & 0xFFu, shift, s_out);
        r.w = decode1(wj >> 24, shift, s_out);
        out4[wbase + j] = r;               // global_store_b128
      }
      if (cn >= nchunks) break;
      c = cn;
      buf ^= 1;
    }
  }
#else
  const long long nchunks = 0;
#endif
  // remainder words + scalar tail: direct path
  const long long gtid = (long long)blockIdx.x * 256 + threadIdx.x;
  const long long gsz = stride * 256;
  for (long long i = (nchunks << 10) + gtid; i < n4; i += gsz) {
    unsigned wj = cw[i];
    float4 r;
    r.x = decode1(wj & 0xFFu, shift, s_out);
    r.y = decode1((wj >> 8) & 0xFFu, shift, s_out);
    r.z = decode1((wj >> 16) & 0xFFu, shift, s_out);
    r.w = decode1(wj >> 24, shift, s_out);
    out4[i] = r;
  }
  for (long long i = n4 * 4 + gtid; i < n; i += gsz)
    out[i] = decode1((unsigned)codes[i], shift, s_out);
}

// ---------------- Pass 2, fallback ----------------
// Recompute codes from x; x staged through LDS with the async pipeline.
__global__ __launch_bounds__(256) void dequant_kernel(
    const float* __restrict__ x, const float* __restrict__ s_ptr,
    const int* __restrict__ params, float* __restrict__ out,
    long long n4, long long n) {
#if defined(__gfx1250__)
  __shared__ float4 tile[2][256];
#endif
  const float s = *s_ptr;
  const float s_out = s * 255.0f;
  const int shift = 8 - params[0] - params[1];
  const float4* __restrict__ x4 = (const float4*)x;
  float4* __restrict__ out4 = (float4*)out;
  const long long stride = gridDim.x;
#if defined(__gfx1250__)
  const long long nchunks = n4 >> 8;      // chunk = 256 float4 (4 KB)
  long long c = blockIdx.x;
  if (c < nchunks) {
    {
      unsigned laddr = (unsigned)(uintptr_t)(void*)&tile[0][threadIdx.x];
      unsigned long long gaddr = (unsigned long long)(x4 + c * 256 + threadIdx.x);
      asm volatile("global_load_async_to_lds_b128 %0, %1, off"
                   :: "v"(laddr), "v"(gaddr) : "memory");
    }
    int buf = 0;
    for (;;) {
      long long cn = c + stride;
      if (cn < nchunks) {
        unsigned laddr = (unsigned)(uintptr_t)(void*)&tile[buf ^ 1][threadIdx.x];
        unsigned long long gaddr = (unsigned long long)(x4 + cn * 256 + threadIdx.x);
        asm volatile("global_load_async_to_lds_b128 %0, %1, off"
                     :: "v"(laddr), "v"(gaddr) : "memory");
        asm volatile("s_wait_asynccnt 0x1" ::: "memory");
      } else {
        asm volatile("s_wait_asynccnt 0x0" ::: "memory");
      }
      float4 v = tile[buf][threadIdx.x];
      float4 r;
      r.x = dequant1(v.x, s, shift, s_out);
      r.y = dequant1(v.y, s, shift, s_out);
      r.z = dequant1(v.z, s, shift, s_out);
      r.w = dequant1(v.w, s, shift, s_out);
      out4[c * 256 + threadIdx.x] = r;
      if (cn >= nchunks) break;
      c = cn;
      buf ^= 1;
    }
  }
#else
  const long long nchunks = 0;
#endif
  const long long gtid = (long long)blockIdx.x * 256 + threadIdx.x;
  const long long gsz = stride * 256;
  for (long long i = (nchunks << 8) + gtid; i < n4; i += gsz) {
    float4 v = x4[i];
    float4 r;
    r.x = dequant1(v.x, s, shift, s_out);
    r.y = dequant1(v.y, s, shift, s_out);
    r.z = dequant1(v.z, s, shift, s_out);
    r.w = dequant1(v.w, s, shift, s_out);
    out4[i] = r;
  }
  for (long long i = n4 * 4 + gtid; i < n; i += gsz)
    out[i] = dequant1(x[i], s, shift, s_out);
}

extern "C" void kernel_launch(void* const* d_in, const int* in_sizes, int n_in,
                              void* d_out, int out_size, void* d_ws, size_t ws_size,
                              hipStream_t stream) {
  const float* x = (const float*)d_in[0];
  const float* s = (const float*)d_in[1];
  const long long n = (long long)in_sizes[0];   // 8*12*512*512 = 25,165,824
  float* out = (float*)d_out;                   // [n] quantized map + [1] s_out
  unsigned long long* mask = (unsigned long long*)d_ws;
  int* params = (int*)((char*)d_ws + 8);
  unsigned char* codes = (unsigned char*)d_ws + 256;  // 16B-aligned u8 codes
  const bool use_codes = (ws_size >= (size_t)n + 512);

  init_mask_kernel<<<1, 1, 0, stream>>>(mask);

  const long long n4 = n >> 2;
  long long want = (n4 + 255) / 256;
  if (want < 1) want = 1;
  const int grid1 = (int)(want < 8192 ? want : 8192);
  if (use_codes)
    quant_mask_store_kernel<<<grid1, 256, 0, stream>>>(x, s, mask, codes, n4, n);
  else
    quant_mask_kernel<<<grid1, 256, 0, stream>>>(x, s, mask, n4, n);

  finalize_kernel<<<1, 1, 0, stream>>>(mask, s, params, out + n);

  if (use_codes) {
    long long nch = (n4 >> 10);
    if (nch < 1) nch = 1;
    const int grid2 = (int)(nch < 4096 ? nch : 4096);
    dequant_codes_kernel<<<grid2, 256, 0, stream>>>(codes, s, params, out, n4, n);
  } else {
    long long nch = (n4 >> 8);
    if (nch < 1) nch = 1;
    const int grid2 = (int)(nch < 4096 ? nch : 4096);
    dequant_kernel<<<grid2, 256, 0, stream>>>(x, s, params, out, n4, n);
  }
}